// ExternalInteraction_3942779977809
// MI455X (gfx1250) — compile-verified
//
#include <hip/hip_runtime.h>
#include <cstdint>
#include <cstddef>

// ---------------------------------------------------------------------------
// ExternalInteraction fused kernel for MI455X (gfx1250).
//   img_sum[b,d]  = sum_i img[b,i,d]     (WMMA f32 16x16x4, A = ones)
//   user_sum[b,d] = sum_u user[b,u,d]
//   out_user[b,u,d] = user[b,u,d] * img_sum[b,d]
//   out_img [b,i,d] = img [b,i,d] * user_sum[b,d]
// Tiles staged HBM->LDS once via GLOBAL_LOAD_ASYNC_TO_LDS_B128 (ASYNCcnt).
// ---------------------------------------------------------------------------

typedef float v2f __attribute__((ext_vector_type(2)));
typedef float v8f __attribute__((ext_vector_type(8)));

#define HAS_WMMA_F32X4 __has_builtin(__builtin_amdgcn_wmma_f32_16x16x4_f32)

namespace {
constexpr int kB = 64;
constexpr int kU = 256;
constexpr int kI = 1024;
constexpr int kD = 512;
constexpr int kDCH = 16;                    // columns per block == WMMA N
constexpr int kThreads = 128;               // 4 wave32
constexpr int kWaves = 4;
constexpr int kChunksPerBatch = kD / kDCH;  // 32
}  // namespace

__device__ __forceinline__ void async_tile_load_b128(uint32_t lds_byte_off,
                                                     const void* gaddr) {
  // CDNA5: copies 16 B from global vaddr into LDS at offset vdst; tracked by ASYNCcnt.
  asm volatile("global_load_async_to_lds_b128 %0, %1, off"
               :
               : "v"(lds_byte_off), "v"(gaddr)
               : "memory");
}

__device__ __forceinline__ void wait_async_zero() {
#if __has_builtin(__builtin_amdgcn_s_wait_asynccnt)
  __builtin_amdgcn_s_wait_asynccnt(0);
#else
  asm volatile("s_wait_asynccnt 0" ::: "memory");
#endif
}

__global__ __launch_bounds__(kThreads) void
interaction_fused_kernel(const float* __restrict__ user,
                         const float* __restrict__ img,
                         float* __restrict__ out_user,
                         float* __restrict__ out_img) {
  __shared__ float s_img[kI * kDCH];        // 64 KB image tile
  __shared__ float s_usr[kU * kDCH];        // 16 KB user tile
  __shared__ float s_part_img[kWaves * 32]; // per-wave partial column sums
  __shared__ float s_part_usr[kWaves * 32];

  const int tid  = threadIdx.x;
  const int lane = tid & 31;
  const int wave = tid >> 5;
  const int blk  = blockIdx.x;
  const int b    = blk / kChunksPerBatch;
  const int dbase = (blk % kChunksPerBatch) * kDCH;

  const float* gimg = img  + ((size_t)b * kI) * kD + dbase;
  const float* gusr = user + ((size_t)b * kU) * kD + dbase;

  // ---- Phase 1: async-stage both tiles into LDS (strided 2D gather) ----
  {
    const uint32_t lds_img0 = (uint32_t)(uintptr_t)&s_img[0];
    for (int idx = tid; idx < kI * 4; idx += kThreads) {   // 4 x b128 per row
      const int row = idx >> 2, ch = idx & 3;
      async_tile_load_b128(lds_img0 + (uint32_t)(row * kDCH + ch * 4) * 4u,
                           (const void*)((const char*)(gimg + (size_t)row * kD) + ch * 16));
    }
    const uint32_t lds_usr0 = (uint32_t)(uintptr_t)&s_usr[0];
    for (int idx = tid; idx < kU * 4; idx += kThreads) {
      const int row = idx >> 2, ch = idx & 3;
      async_tile_load_b128(lds_usr0 + (uint32_t)(row * kDCH + ch * 4) * 4u,
                           (const void*)((const char*)(gusr + (size_t)row * kD) + ch * 16));
    }
  }
  wait_async_zero();
  __syncthreads();

  // ---- Phase 2: column sums. Each wave reduces a quarter of the rows. ----
  const int half = lane >> 4;   // which half of the wave
  const int n    = lane & 15;   // column within the 16-wide chunk
  const int q    = wave;        // row-quarter index

#if HAS_WMMA_F32X4
  // A = ones(16x4)  =>  D[m,n] = sum_k B[k,n]; every lane's c[0] == colsum(n).
  v2f ones; ones[0] = 1.0f; ones[1] = 1.0f;
  v8f c_img = {0.f, 0.f, 0.f, 0.f, 0.f, 0.f, 0.f, 0.f};
  v8f c_usr = {0.f, 0.f, 0.f, 0.f, 0.f, 0.f, 0.f, 0.f};
  for (int rg = q * (kI / kWaves); rg < (q + 1) * (kI / kWaves); rg += 4) {
    v2f bv;
    bv[0] = s_img[(rg + 2 * half    ) * kDCH + n];
    bv[1] = s_img[(rg + 2 * half + 1) * kDCH + n];
    c_img = __builtin_amdgcn_wmma_f32_16x16x4_f32(false, ones, false, bv,
                                                  (short)0, c_img, false, false);
  }
  for (int rg = q * (kU / kWaves); rg < (q + 1) * (kU / kWaves); rg += 4) {
    v2f bv;
    bv[0] = s_usr[(rg + 2 * half    ) * kDCH + n];
    bv[1] = s_usr[(rg + 2 * half + 1) * kDCH + n];
    c_usr = __builtin_amdgcn_wmma_f32_16x16x4_f32(false, ones, false, bv,
                                                  (short)0, c_usr, false, false);
  }
  if (lane < 16) {
    s_part_img[wave * 16 + n] = c_img[0];
    s_part_usr[wave * 16 + n] = c_usr[0];
  }
  __syncthreads();
#else
  // VALU fallback: each (half, n) lane sums every other row of its quarter.
  float acc_i = 0.f, acc_u = 0.f;
  for (int r = q * (kI / kWaves) + half; r < (q + 1) * (kI / kWaves); r += 2)
    acc_i += s_img[r * kDCH + n];
  for (int r = q * (kU / kWaves) + half; r < (q + 1) * (kU / kWaves); r += 2)
    acc_u += s_usr[r * kDCH + n];
  s_part_img[wave * 32 + lane] = acc_i;
  s_part_usr[wave * 32 + lane] = acc_u;
  __syncthreads();
#endif

  // ---- Combine partials: each lane needs sums for its 4 columns. ----
  const int c0 = (lane & 3) * 4;   // 4 consecutive columns per lane
  float tI[4], tU[4];
  for (int j = 0; j < 4; ++j) {
    const int col = c0 + j;
    float si = 0.f, su = 0.f;
    for (int w = 0; w < kWaves; ++w) {
#if HAS_WMMA_F32X4
      si += s_part_img[w * 16 + col];
      su += s_part_usr[w * 16 + col];
#else
      si += s_part_img[w * 32 + col] + s_part_img[w * 32 + 16 + col];
      su += s_part_usr[w * 32 + col] + s_part_usr[w * 32 + 16 + col];
#endif
    }
    tI[j] = si;
    tU[j] = su;
  }

  // ---- Phase 3: broadcast multiply, float4 LDS reads + coalesced stores ----
  const int rsub = lane >> 2;   // 8 rows covered per iteration (4 lanes/row)
  float* ou = out_user + ((size_t)b * kU) * kD + dbase;
  float* oi = out_img  + ((size_t)b * kI) * kD + dbase;

  for (int r0 = q * (kU / kWaves); r0 < (q + 1) * (kU / kWaves); r0 += 8) {
    const int r = r0 + rsub;
    const float4 v = *(const float4*)&s_usr[r * kDCH + c0];
    float4 o;
    o.x = v.x * tI[0]; o.y = v.y * tI[1]; o.z = v.z * tI[2]; o.w = v.w * tI[3];
    *(float4*)(ou + (size_t)r * kD + c0) = o;
  }
  for (int r0 = q * (kI / kWaves); r0 < (q + 1) * (kI / kWaves); r0 += 8) {
    const int r = r0 + rsub;
    const float4 v = *(const float4*)&s_img[r * kDCH + c0];
    float4 o;
    o.x = v.x * tU[0]; o.y = v.y * tU[1]; o.z = v.z * tU[2]; o.w = v.w * tU[3];
    *(float4*)(oi + (size_t)r * kD + c0) = o;
  }
}

extern "C" void kernel_launch(void* const* d_in, const int* in_sizes, int n_in,
                              void* d_out, int out_size, void* d_ws, size_t ws_size,
                              hipStream_t stream) {
  (void)in_sizes; (void)n_in; (void)out_size; (void)d_ws; (void)ws_size;
  const float* user = (const float*)d_in[0];   // [64,256,512] f32
  const float* img  = (const float*)d_in[1];   // [64,1024,512] f32
  float* out_user = (float*)d_out;                              // [64,256,512]
  float* out_img  = out_user + (size_t)kB * kU * kD;            // [64,1024,512]

  dim3 grid(kB * kChunksPerBatch);   // 2048 blocks: (batch, 16-col chunk)
  dim3 block(kThreads);              // 4 wave32
  interaction_fused_kernel<<<grid, block, 0, stream>>>(user, img, out_user, out_img);
}